// HyperScatteringModule_56203942036042
// MI455X (gfx1250) — compile-verified
//
#include <hip/hip_runtime.h>

#define NUM_V 65536
#define NUM_E 16384
#define NNZ   1048576
#define FEAT  128

typedef __attribute__((ext_vector_type(2))) float v2f;
typedef __attribute__((ext_vector_type(8))) float v8f;

// ---------------------------------------------------------------- utilities

__global__ void zero_i32(int* __restrict__ p, int n) {
  int i = blockIdx.x * blockDim.x + threadIdx.x;
  int st = gridDim.x * blockDim.x;
  for (; i < n; i += st) p[i] = 0;
}

__global__ void copy_i32(const int* __restrict__ s, int* __restrict__ d, int n) {
  int i = blockIdx.x * blockDim.x + threadIdx.x;
  int st = gridDim.x * blockDim.x;
  for (; i < n; i += st) d[i] = s[i];
}

__global__ void copy_f4(const float4* __restrict__ s, float4* __restrict__ d, int n) {
  int i = blockIdx.x * blockDim.x + threadIdx.x;
  int st = gridDim.x * blockDim.x;
  for (; i < n; i += st) d[i] = s[i];
}

// ----------------------------------------------------- degree / CSR building

__global__ void count_deg(const int* __restrict__ vid, const int* __restrict__ eid,
                          int* __restrict__ cnt_v, int* __restrict__ cnt_e) {
  int i = blockIdx.x * blockDim.x + threadIdx.x;
  int st = gridDim.x * blockDim.x;
  for (; i < NNZ; i += st) {
    atomicAdd(&cnt_v[vid[i]], 1);
    atomicAdd(&cnt_e[eid[i]], 1);
  }
}

__global__ void make_inv(const int* __restrict__ cnt, float* __restrict__ inv, int n) {
  int i = blockIdx.x * blockDim.x + threadIdx.x;
  int st = gridDim.x * blockDim.x;
  for (; i < n; i += st) {
    int c = cnt[i];
    inv[i] = (c > 0) ? (1.0f / (float)c) : 0.0f;
  }
}

// single-block Hillis-Steele exclusive scan over n counts; writes off[0..n]
__global__ void scan_excl(const int* __restrict__ cnt, int* __restrict__ off, int n) {
  __shared__ int tmp[1024];
  __shared__ int carry_s;
  if (threadIdx.x == 0) carry_s = 0;
  __syncthreads();
  for (int base = 0; base < n; base += 1024) {
    int i = base + (int)threadIdx.x;
    int val = (i < n) ? cnt[i] : 0;
    int sum = val;
    tmp[threadIdx.x] = val;
    __syncthreads();
    for (int s = 1; s < 1024; s <<= 1) {
      int t = (threadIdx.x >= (unsigned)s) ? tmp[threadIdx.x - s] : 0;
      __syncthreads();
      sum += t;
      tmp[threadIdx.x] = sum;
      __syncthreads();
    }
    if (i < n) off[i] = carry_s + sum - val;
    __syncthreads();
    if (threadIdx.x == 1023) carry_s += tmp[1023];
    __syncthreads();
  }
  if (threadIdx.x == 0) off[n] = carry_s;
}

__global__ void csr_fill(const int* __restrict__ vid, const int* __restrict__ eid,
                         int* __restrict__ pos_e, int* __restrict__ pos_v,
                         int* __restrict__ csr_e_vid, int* __restrict__ csr_v_eid) {
  int i = blockIdx.x * blockDim.x + threadIdx.x;
  int st = gridDim.x * blockDim.x;
  for (; i < NNZ; i += st) {
    int v = vid[i], e = eid[i];
    int pe = atomicAdd(&pos_e[e], 1);
    csr_e_vid[pe] = v;
    int pv = atomicAdd(&pos_v[v], 1);
    csr_v_eid[pv] = e;
  }
}

// ------------------------------------------------------ diffusion hot loops
// one wave (32 lanes) per row; each lane owns a float4 slice of the 512B row

__global__ void v2e_gather(const float* __restrict__ x,
                           const float* __restrict__ inv_v,
                           const float* __restrict__ inv_e,
                           const int* __restrict__ off_e,
                           const int* __restrict__ csr_e_vid,
                           float* __restrict__ ef) {
  const int lane = (int)(threadIdx.x & 31u);
  int wave = (int)((blockIdx.x * blockDim.x + threadIdx.x) >> 5);
  const int nwaves = (int)((gridDim.x * blockDim.x) >> 5);
  for (int e = wave; e < NUM_E; e += nwaves) {
    const int beg = __builtin_amdgcn_readfirstlane(off_e[e]);
    const int end = __builtin_amdgcn_readfirstlane(off_e[e + 1]);
    float4 acc = make_float4(0.f, 0.f, 0.f, 0.f);
    for (int i = beg; i < end; ++i) {
      const int v = __builtin_amdgcn_readfirstlane(csr_e_vid[i]);
      if (i + 1 < end) {
        const int vn = __builtin_amdgcn_readfirstlane(csr_e_vid[i + 1]);
        __builtin_prefetch(x + (size_t)vn * FEAT + (size_t)lane * 4, 0, 1);
      }
      const float wv = inv_v[v];
      const float4 xx = ((const float4*)(x + (size_t)v * FEAT))[lane];
      acc.x += xx.x * wv;
      acc.y += xx.y * wv;
      acc.z += xx.z * wv;
      acc.w += xx.w * wv;
    }
    const float se = inv_e[e];
    float4 r;
    r.x = acc.x * se; r.y = acc.y * se; r.z = acc.z * se; r.w = acc.w * se;
    ((float4*)(ef + (size_t)e * FEAT))[lane] = r;
  }
}

__global__ void e2v_gather(const float* __restrict__ ef,
                           const int* __restrict__ off_v,
                           const int* __restrict__ csr_v_eid,
                           float* __restrict__ xout) {
  const int lane = (int)(threadIdx.x & 31u);
  int wave = (int)((blockIdx.x * blockDim.x + threadIdx.x) >> 5);
  const int nwaves = (int)((gridDim.x * blockDim.x) >> 5);
  for (int v = wave; v < NUM_V; v += nwaves) {
    const int beg = __builtin_amdgcn_readfirstlane(off_v[v]);
    const int end = __builtin_amdgcn_readfirstlane(off_v[v + 1]);
    float4 acc = make_float4(0.f, 0.f, 0.f, 0.f);
    for (int i = beg; i < end; ++i) {
      const int e = __builtin_amdgcn_readfirstlane(csr_v_eid[i]);
      if (i + 1 < end) {
        const int en = __builtin_amdgcn_readfirstlane(csr_v_eid[i + 1]);
        __builtin_prefetch(ef + (size_t)en * FEAT + (size_t)lane * 4, 0, 1);
      }
      const float4 t = ((const float4*)(ef + (size_t)e * FEAT))[lane];
      acc.x += t.x; acc.y += t.y; acc.z += t.z; acc.w += t.w;
    }
    ((float4*)(xout + (size_t)v * FEAT))[lane] = acc;
  }
}

// --------------------------------------------- wavelet row via f32 WMMA
// w[n] = early[n] + cLate * late[n]  for 16 n per WMMA, via
// V_WMMA_F32_16X16X4_F32 with A = one-hot row [1, cLate, 0, 0] (exact in f32).
// A layout: lanes 0-15 hold A[M=lane][K=0..1] in the 2 VGPRs (lane 0 = row 0).
// B layout: lanes 0-15 hold B[K=0][n], B[K=1][n]; lanes 16-31 hold K=2,3 (zero).
// D row 0 lands in VGPR0 lanes 0-15. Writes relu(w) and relu(-w) to d_out.

__global__ void wavelet_finalize(const float* __restrict__ early,
                                 const float* __restrict__ late,
                                 float* __restrict__ out, int row, float cLate) {
  const int lane = (int)(threadIdx.x & 31u);
  const int wave = (int)((blockIdx.x * blockDim.x + threadIdx.x) >> 5);
  const int nwaves = (int)((gridDim.x * blockDim.x) >> 5);
  const int ngroups = (NUM_V * FEAT) / 16;

  v2f a;
  a[0] = (lane == 0) ? 1.0f : 0.0f;
  a[1] = (lane == 0) ? cLate : 0.0f;

  for (int g = wave; g < ngroups; g += nwaves) {
    const int n = g * 16 + lane;
    v2f b;
    b[0] = 0.0f;
    b[1] = 0.0f;
    if (lane < 16) {
      b[0] = early[n];
      b[1] = late[n];
    }
#if __has_builtin(__builtin_amdgcn_wmma_f32_16x16x4_f32)
    v8f c = {};
    v8f d = __builtin_amdgcn_wmma_f32_16x16x4_f32(
        /*neg_a=*/false, a, /*neg_b=*/false, b,
        /*c_mod=*/(short)0, c, /*reuse_a=*/false, /*reuse_b=*/false);
    const float wv = d[0];
#else
    const float wv = b[0] + cLate * b[1];
#endif
    if (lane < 16) {
      const int v = n >> 7;
      const int f = n & (FEAT - 1);
      float* o = out + (size_t)v * (6 * 2 * FEAT) + (size_t)row * (2 * FEAT) + f;
      o[0]    = (wv > 0.0f) ? wv : 0.0f;
      o[FEAT] = (wv < 0.0f) ? -wv : 0.0f;
    }
  }
}

// ------------------------------------------------------------------- driver

extern "C" void kernel_launch(void* const* d_in, const int* in_sizes, int n_in,
                              void* d_out, int out_size, void* d_ws, size_t ws_size,
                              hipStream_t stream) {
  const float* X   = (const float*)d_in[0];
  const int*   vid = (const int*)d_in[1];
  const int*   eid = (const int*)d_in[2];
  float* out = (float*)d_out;

  char* wsp = (char*)d_ws;
  auto carve = [&](size_t bytes) -> char* {
    char* p = wsp;
    wsp += (bytes + 255) & ~(size_t)255;
    return p;
  };
  float* xA        = (float*)carve((size_t)NUM_V * FEAT * 4);
  float* xB        = (float*)carve((size_t)NUM_V * FEAT * 4);
  float* hold      = (float*)carve((size_t)NUM_V * FEAT * 4);
  float* ef        = (float*)carve((size_t)NUM_E * FEAT * 4);
  int*   csr_e_vid = (int*)carve((size_t)NNZ * 4);
  int*   csr_v_eid = (int*)carve((size_t)NNZ * 4);
  int*   cnt_v     = (int*)carve((size_t)NUM_V * 4);
  int*   cnt_e     = (int*)carve((size_t)NUM_E * 4);
  int*   off_v     = (int*)carve((size_t)(NUM_V + 1) * 4);
  int*   off_e     = (int*)carve((size_t)(NUM_E + 1) * 4);
  int*   pos_v     = (int*)carve((size_t)NUM_V * 4);
  int*   pos_e     = (int*)carve((size_t)NUM_E * 4);
  float* inv_v     = (float*)carve((size_t)NUM_V * 4);
  float* inv_e     = (float*)carve((size_t)NUM_E * 4);

  // one-time structure build (re-done every launch: deterministic, no state)
  zero_i32<<<256, 256, 0, stream>>>(cnt_v, NUM_V);
  zero_i32<<<64, 256, 0, stream>>>(cnt_e, NUM_E);
  count_deg<<<2048, 256, 0, stream>>>(vid, eid, cnt_v, cnt_e);
  make_inv<<<256, 256, 0, stream>>>(cnt_v, inv_v, NUM_V);
  make_inv<<<64, 256, 0, stream>>>(cnt_e, inv_e, NUM_E);
  scan_excl<<<1, 1024, 0, stream>>>(cnt_v, off_v, NUM_V);
  scan_excl<<<1, 1024, 0, stream>>>(cnt_e, off_e, NUM_E);
  copy_i32<<<256, 256, 0, stream>>>(off_v, pos_v, NUM_V);
  copy_i32<<<64, 256, 0, stream>>>(off_e, pos_e, NUM_E);
  csr_fill<<<2048, 256, 0, stream>>>(vid, eid, pos_e, pos_v, csr_e_vid, csr_v_eid);

  // 16 diffusion rounds with incremental wavelet emission
  const float* src = X;
  float* bufs[2] = {xA, xB};
  int cur = 0;
  const int FG = 4096, FB = 256;     // finalize launch: 32768 waves / 512K groups
  const int NF4 = NUM_V * FEAT / 4;  // float4 count for hold copies
  for (int k = 1; k <= 16; ++k) {
    float* dst = bufs[cur];
    v2e_gather<<<2048, 256, 0, stream>>>(src, inv_v, inv_e, off_e, csr_e_vid, ef);
    e2v_gather<<<8192, 256, 0, stream>>>(ef, off_v, csr_v_eid, dst);
    if (k == 1) {  // wav0 = L0 - L1
      wavelet_finalize<<<FG, FB, 0, stream>>>(X, dst, out, 0, -1.0f);
    } else if (k == 2) {  // wav1 = L1 - L2 ; hold <- L2
      wavelet_finalize<<<FG, FB, 0, stream>>>(src, dst, out, 1, -1.0f);
      copy_f4<<<2048, 256, 0, stream>>>((const float4*)dst, (float4*)hold, NF4);
    } else if (k == 4) {  // wav2 = L2 - L4 ; hold <- L4
      wavelet_finalize<<<FG, FB, 0, stream>>>(hold, dst, out, 2, -1.0f);
      copy_f4<<<2048, 256, 0, stream>>>((const float4*)dst, (float4*)hold, NF4);
    } else if (k == 8) {  // wav3 = L4 - L8 ; hold <- L8
      wavelet_finalize<<<FG, FB, 0, stream>>>(hold, dst, out, 3, -1.0f);
      copy_f4<<<2048, 256, 0, stream>>>((const float4*)dst, (float4*)hold, NF4);
    } else if (k == 16) {  // wav4 = L8 - L16 ; wav5 = L16
      wavelet_finalize<<<FG, FB, 0, stream>>>(hold, dst, out, 4, -1.0f);
      wavelet_finalize<<<FG, FB, 0, stream>>>(dst, dst, out, 5, 0.0f);
    }
    src = dst;
    cur ^= 1;
  }
  (void)in_sizes; (void)n_in; (void)out_size; (void)ws_size;
}